// TAM_21328807592452
// MI455X (gfx1250) — compile-verified
//
#include <hip/hip_runtime.h>
#include <math.h>

// ---------------- problem constants ----------------
constexpr int BN  = 128;   // batch
constexpr int CC  = 8;     // channels
constexpr int FF  = 58;    // freq
constexpr int TT  = 249;   // time
constexpr int RC  = 24;    // residual channels
constexpr int HIN = FF + 2;   // 60
constexpr int WIN = TT + 2;   // 251

constexpr long long PLANE = (long long)FF * TT;       // 14442
constexpr long long CB    = (long long)CC * PLANE;    // 115536  per-batch Q/K/V block
constexpr long long NBUF  = (long long)BN * CB;       // full buffer elements

constexpr int CT = CC * TT;   // 1992  (freq-attn d)
constexpr int FT = (int)PLANE;// 14442 (chan-attn d)
constexpr int CF = CC * FF;   // 464   (time-attn d)
constexpr int L2OUT = (CF - 2) / 2 + 1; // 232 (ks2 stride2 conv output)

typedef float v2f __attribute__((ext_vector_type(2)));
typedef float v8f __attribute__((ext_vector_type(8)));

__device__ __forceinline__ v2f load2(const float* p) {
    v2f v;
    __builtin_memcpy(&v, p, 2 * sizeof(float));   // align-4 b64 load
    return v;
}

__device__ __forceinline__ v8f wmma4(v2f a, v2f b, v8f c) {
    return __builtin_amdgcn_wmma_f32_16x16x4_f32(false, a, false, b, (short)0, c, false, false);
}

// =====================================================================
// fp32 WMMA GEMM, 32x64 tile per wave (2x4 accumulators, 8 WMMA / K-step
// from 2 A-fragment + 4 B-fragment loads). All fragment pointers are
// scalar variables (not arrays) so clang keeps GLOBAL address space
// inference -> global_load_b64, not flat_load_b64.
//   C[b] (M x N) = scale * A[b] * B[b](^T) (+bias)(+= C)
//   A element (row,k) = A[aBatch*b + row*lda + k*aKs + aKo]
//   B element (k,col) = TRANSB ? B[bBatch*b + col*ldb + k]
//                              : B[bBatch*b + k*ldb + col*bCs + bCo]
// Edge policy: row/col indices clamped (loads stay in-bounds); garbage
// products land only in never-stored accumulator slots. Only the K-tail
// needs zero-selects (branchless v_cndmask).
// =====================================================================
template <bool TRANSB, bool AUNIT>
__global__ __launch_bounds__(32)
void gemm_f32_wmma(const float* __restrict__ A, long long aBatch, int lda, int aKs, int aKo,
                   const float* __restrict__ Bm, long long bBatch, int ldb, int bCs, int bCo,
                   float* __restrict__ Cm, long long cBatch, int ldc,
                   int M, int N, int K,
                   float scale, const float* __restrict__ bias, int accumulate)
{
    const int b    = blockIdx.z;
    const int lane = threadIdx.x;
    const int hl   = lane >> 4;      // 0/1: which K-pair this lane supplies
    const int l16  = lane & 15;

    const float* Ab = A  + aBatch * b;
    const float* Bb = Bm + bBatch * b;
    float*       Cb = Cm + cBatch * b;

    const int rowBase = blockIdx.y * 32;
    const int colBase = blockIdx.x * 64;

    // clamped fragment rows / cols (always in-bounds)
    const int r0 = min(rowBase + l16,      M - 1);
    const int r1 = min(rowBase + 16 + l16, M - 1);
    const int c0 = min(colBase + l16,      N - 1);
    const int c1 = min(colBase + 16 + l16, N - 1);
    const int c2 = min(colBase + 32 + l16, N - 1);
    const int c3 = min(colBase + 48 + l16, N - 1);

    const float* pa0 = Ab + (long long)r0 * lda + (long long)(hl * 2) * aKs + aKo;
    const float* pa1 = Ab + (long long)r1 * lda + (long long)(hl * 2) * aKs + aKo;
    const float* pb0;
    const float* pb1;
    const float* pb2;
    const float* pb3;
    long long bStep, bEs;
    if (TRANSB) {
        pb0 = Bb + (long long)c0 * ldb + hl * 2;
        pb1 = Bb + (long long)c1 * ldb + hl * 2;
        pb2 = Bb + (long long)c2 * ldb + hl * 2;
        pb3 = Bb + (long long)c3 * ldb + hl * 2;
        bStep = 4; bEs = 1;
    } else {
        const float* brow = Bb + (long long)(hl * 2) * ldb + bCo;
        pb0 = brow + (long long)c0 * bCs;
        pb1 = brow + (long long)c1 * bCs;
        pb2 = brow + (long long)c2 * bCs;
        pb3 = brow + (long long)c3 * bCs;
        bStep = 4LL * ldb; bEs = ldb;
    }
    const long long aStep = 4LL * aKs;

    v8f acc00 = {}, acc01 = {}, acc02 = {}, acc03 = {};
    v8f acc10 = {}, acc11 = {}, acc12 = {}, acc13 = {};

    const int K4 = K & ~3;
#pragma unroll 2
    for (int k0 = 0; k0 < K4; k0 += 4) {
        v2f a0, a1, b0, b1, b2, b3;
        if (AUNIT) { a0 = load2(pa0); a1 = load2(pa1); }
        else       { a0.x = pa0[0]; a0.y = pa0[aKs]; a1.x = pa1[0]; a1.y = pa1[aKs]; }
        if (TRANSB) {
            b0 = load2(pb0); b1 = load2(pb1); b2 = load2(pb2); b3 = load2(pb3);
        } else {
            b0.x = pb0[0]; b0.y = pb0[bEs];
            b1.x = pb1[0]; b1.y = pb1[bEs];
            b2.x = pb2[0]; b2.y = pb2[bEs];
            b3.x = pb3[0]; b3.y = pb3[bEs];
        }

        acc00 = wmma4(a0, b0, acc00);
        acc01 = wmma4(a0, b1, acc01);
        acc02 = wmma4(a0, b2, acc02);
        acc03 = wmma4(a0, b3, acc03);
        acc10 = wmma4(a1, b0, acc10);
        acc11 = wmma4(a1, b1, acc11);
        acc12 = wmma4(a1, b2, acc12);
        acc13 = wmma4(a1, b3, acc13);

        pa0 += aStep; pa1 += aStep;
        pb0 += bStep; pb1 += bStep; pb2 += bStep; pb3 += bStep;
    }

    // ---- K tail (1..3 leftover): clamped loads + branchless zero-select ----
    if (K4 < K) {
        const int ka  = K4 + hl * 2;
        const bool vx = ka < K, vy = (ka + 1) < K;
        const long long kx = (long long)min(ka,     K - 1);
        const long long ky = (long long)min(ka + 1, K - 1);

        v2f a0, a1, b0, b1, b2, b3;
        a0.x = Ab[(long long)r0 * lda + kx * aKs + aKo];
        a0.y = Ab[(long long)r0 * lda + ky * aKs + aKo];
        a1.x = Ab[(long long)r1 * lda + kx * aKs + aKo];
        a1.y = Ab[(long long)r1 * lda + ky * aKs + aKo];
        if (TRANSB) {
            b0.x = Bb[(long long)c0 * ldb + kx];  b0.y = Bb[(long long)c0 * ldb + ky];
            b1.x = Bb[(long long)c1 * ldb + kx];  b1.y = Bb[(long long)c1 * ldb + ky];
            b2.x = Bb[(long long)c2 * ldb + kx];  b2.y = Bb[(long long)c2 * ldb + ky];
            b3.x = Bb[(long long)c3 * ldb + kx];  b3.y = Bb[(long long)c3 * ldb + ky];
        } else {
            const long long ci0 = (long long)c0 * bCs + bCo, ci1 = (long long)c1 * bCs + bCo;
            const long long ci2 = (long long)c2 * bCs + bCo, ci3 = (long long)c3 * bCs + bCo;
            b0.x = Bb[kx * ldb + ci0];  b0.y = Bb[ky * ldb + ci0];
            b1.x = Bb[kx * ldb + ci1];  b1.y = Bb[ky * ldb + ci1];
            b2.x = Bb[kx * ldb + ci2];  b2.y = Bb[ky * ldb + ci2];
            b3.x = Bb[kx * ldb + ci3];  b3.y = Bb[ky * ldb + ci3];
        }
        a0.x = vx ? a0.x : 0.f; a0.y = vy ? a0.y : 0.f;
        a1.x = vx ? a1.x : 0.f; a1.y = vy ? a1.y : 0.f;
        b0.x = vx ? b0.x : 0.f; b0.y = vy ? b0.y : 0.f;
        b1.x = vx ? b1.x : 0.f; b1.y = vy ? b1.y : 0.f;
        b2.x = vx ? b2.x : 0.f; b2.y = vy ? b2.y : 0.f;
        b3.x = vx ? b3.x : 0.f; b3.y = vy ? b3.y : 0.f;

        acc00 = wmma4(a0, b0, acc00);
        acc01 = wmma4(a0, b1, acc01);
        acc02 = wmma4(a0, b2, acc02);
        acc03 = wmma4(a0, b3, acc03);
        acc10 = wmma4(a1, b0, acc10);
        acc11 = wmma4(a1, b1, acc11);
        acc12 = wmma4(a1, b2, acc12);
        acc13 = wmma4(a1, b3, acc13);
    }

    // ---- store subtiles (D: vgpr r -> row rB + r + hl*8, lane l16 -> col) ----
    auto store_tile = [&](const v8f& acc, int rB, int cB) {
        const int col = cB + l16;
        if (col < N) {
#pragma unroll
            for (int r = 0; r < 8; ++r) {
                const int row = rB + r + hl * 8;
                if (row < M) {
                    float v = acc[r] * scale;
                    if (bias) v += bias[row];
                    const long long idx = (long long)row * ldc + col;
                    Cb[idx] = accumulate ? (Cb[idx] + v) : v;
                }
            }
        }
    };
    store_tile(acc00, rowBase,      colBase);
    store_tile(acc01, rowBase,      colBase + 16);
    store_tile(acc02, rowBase,      colBase + 32);
    store_tile(acc03, rowBase,      colBase + 48);
    store_tile(acc10, rowBase + 16, colBase);
    store_tile(acc11, rowBase + 16, colBase + 16);
    store_tile(acc12, rowBase + 16, colBase + 32);
    store_tile(acc13, rowBase + 16, colBase + 48);
}

// =====================================================================
// Column softmax (softmax over axis=1 / row index, per column)
// =====================================================================
__global__ __launch_bounds__(32)
void softmax_col(float* __restrict__ S, long long sBatch, int ld, int rows, int cols)
{
    const int m = blockIdx.x;
    if (m >= cols) return;
    float* p = S + sBatch * blockIdx.y + m;
    const int lane = threadIdx.x;

    float mx = -3.4e38f;
    for (int n = lane; n < rows; n += 32) mx = fmaxf(mx, p[(long long)n * ld]);
    for (int o = 16; o > 0; o >>= 1)      mx = fmaxf(mx, __shfl_xor(mx, o, 32));

    float sum = 0.f;
    for (int n = lane; n < rows; n += 32) {
        float e = __expf(p[(long long)n * ld] - mx);
        p[(long long)n * ld] = e;
        sum += e;
    }
    for (int o = 16; o > 0; o >>= 1) sum += __shfl_xor(sum, o, 32);

    const float inv = 1.0f / sum;
    for (int n = lane; n < rows; n += 32) p[(long long)n * ld] *= inv;
}

// =====================================================================
__global__ void differ_kernel(const float* __restrict__ x, const float* __restrict__ xconv,
                              float* __restrict__ differ, long long total)
{
    long long i = (long long)blockIdx.x * blockDim.x + threadIdx.x;
    if (i >= total) return;
    int t = (int)(i % TT);
    differ[i] = (t == 0) ? x[i] : (xconv[i] - x[i - 1]);
}

// =====================================================================
// residual: 3x3 VALID conv + BN, plus per-(b,c) plane-sum into theta
// =====================================================================
__global__ __launch_bounds__(256)
void resconv_kernel(const float* __restrict__ inp, const float* __restrict__ w,
                    const float* __restrict__ br,
                    const float* __restrict__ gam, const float* __restrict__ bet,
                    const float* __restrict__ mn,  const float* __restrict__ vr,
                    float* __restrict__ xr, float* __restrict__ theta)
{
    __shared__ float red[256];
    const int c = blockIdx.y;
    const int b = blockIdx.z;
    const int i = blockIdx.x * 256 + threadIdx.x;

    const float g  = gam[c] * rsqrtf(vr[c] + 1e-5f);
    const float sh = bet[c] - mn[c] * g;

    float val = 0.f;
    if (i < (int)PLANE) {
        const int f = i / TT, t = i - f * TT;
        const float* ip = inp + (long long)b * HIN * WIN + (long long)f * WIN + t;
        const float* wc = w + c * 9;
        float a = br[c];
        a += wc[0] * ip[0]         + wc[1] * ip[1]           + wc[2] * ip[2];
        a += wc[3] * ip[WIN]       + wc[4] * ip[WIN + 1]     + wc[5] * ip[WIN + 2];
        a += wc[6] * ip[2 * WIN]   + wc[7] * ip[2 * WIN + 1] + wc[8] * ip[2 * WIN + 2];
        val = a * g + sh;
        xr[((long long)b * RC + c) * PLANE + i] = val;
    }
    red[threadIdx.x] = val;
    __syncthreads();
    for (int s = 128; s > 0; s >>= 1) {
        if (threadIdx.x < s) red[threadIdx.x] += red[threadIdx.x + s];
        __syncthreads();
    }
    if (threadIdx.x == 0) atomicAdd(&theta[b * RC + c], red[0]);
}

// =====================================================================
__global__ __launch_bounds__(128)
void coefs_kernel(const float* __restrict__ theta,
                  const float* __restrict__ w1, const float* __restrict__ b1,
                  const float* __restrict__ w2, const float* __restrict__ b2,
                  float* __restrict__ coefs)
{
    const int b = threadIdx.x;
    if (b >= BN) return;
    float th[RC];
    const float invP = 1.0f / (float)PLANE;
    for (int c = 0; c < RC; ++c) th[c] = theta[b * RC + c] * invP;
    float h[6];
    for (int j = 0; j < 6; ++j) {
        float s = b1[j];
        for (int c = 0; c < RC; ++c) s += w1[j * RC + c] * th[c];
        h[j] = fmaxf(s, 0.f);
    }
    const float lam[4] = {1.f, 1.f, 0.5f, 0.5f};
    const float ini[4] = {1.f, 0.f, 0.f, 0.f};
    for (int k = 0; k < 4 * RC; ++k) {
        float s = b2[k];
        for (int j = 0; j < 6; ++j) s += w2[k * 6 + j] * h[j];
        float v = 2.f / (1.f + __expf(-s)) - 1.f;     // 2*sigmoid - 1
        coefs[b * 4 * RC + k] = v * lam[k & 3] + ini[k & 3];
    }
}

__global__ void zero_kernel(float* p, long long n)
{
    long long i = (long long)blockIdx.x * blockDim.x + threadIdx.x;
    if (i < n) p[i] = 0.f;
}

// =====================================================================
__global__ void final_kernel(float* __restrict__ out, const float* __restrict__ xr,
                             const float* __restrict__ coefs, long long total)
{
    long long i = (long long)blockIdx.x * blockDim.x + threadIdx.x;
    if (i >= total) return;
    const long long plane = i / PLANE;  // b*RC + c
    const float xv = xr[i];
    const float* cf = coefs + plane * 4;
    const float y = fmaxf(fmaf(xv, cf[0], cf[2]), fmaf(xv, cf[1], cf[3]));
    out[i] += y;
}

// ---------------------------------------------------------------------
static inline void launch_gemm(hipStream_t s,
    const float* A, long long aB, int lda, int aKs, int aKo,
    const float* Bm, long long bB, int ldb, int bCs, int bCo, int transB,
    float* C, long long cB, int ldc, int M, int N, int K,
    float scale, const float* bias, int acc, int batches)
{
    dim3 grid((N + 63) / 64, (M + 31) / 32, batches);
    dim3 blk(32, 1, 1);
    if (transB) {
        gemm_f32_wmma<true, true><<<grid, blk, 0, s>>>(A, aB, lda, aKs, aKo,
            Bm, bB, ldb, bCs, bCo, C, cB, ldc, M, N, K, scale, bias, acc);
    } else if (aKs == 1) {
        gemm_f32_wmma<false, true><<<grid, blk, 0, s>>>(A, aB, lda, aKs, aKo,
            Bm, bB, ldb, bCs, bCo, C, cB, ldc, M, N, K, scale, bias, acc);
    } else {
        gemm_f32_wmma<false, false><<<grid, blk, 0, s>>>(A, aB, lda, aKs, aKo,
            Bm, bB, ldb, bCs, bCo, C, cB, ldc, M, N, K, scale, bias, acc);
    }
}

extern "C" void kernel_launch(void* const* d_in, const int* in_sizes, int n_in,
                              void* d_out, int out_size, void* d_ws, size_t ws_size,
                              hipStream_t stream)
{
    (void)in_sizes; (void)n_in; (void)out_size; (void)ws_size;

    const float* x      = (const float*)d_in[0];
    const float* inp    = (const float*)d_in[1];
    const float* w_diff = (const float*)d_in[2];
    const float* b_diff = (const float*)d_in[3];
    const float* w_res  = (const float*)d_in[4];
    const float* b_res  = (const float*)d_in[5];
    const float* bn_g   = (const float*)d_in[6];
    const float* bn_b   = (const float*)d_in[7];
    const float* bn_m   = (const float*)d_in[8];
    const float* bn_v   = (const float*)d_in[9];
    const float* dy_w1  = (const float*)d_in[10];
    const float* dy_b1  = (const float*)d_in[11];
    const float* dy_w2  = (const float*)d_in[12];
    const float* dy_b2  = (const float*)d_in[13];
    const float* wq     = (const float*)d_in[14];
    const float* bq     = (const float*)d_in[15];
    const float* wk     = (const float*)d_in[16];
    const float* bk     = (const float*)d_in[17];
    const float* fq0_w  = (const float*)d_in[18];
    const float* fq0_b  = (const float*)d_in[19];
    const float* fk0_w  = (const float*)d_in[20];
    const float* fk0_b  = (const float*)d_in[21];
    const float* cq0_w  = (const float*)d_in[22];
    const float* cq0_b  = (const float*)d_in[23];
    const float* ck0_w  = (const float*)d_in[24];
    const float* ck0_b  = (const float*)d_in[25];
    const float* tq0_w  = (const float*)d_in[26];
    const float* tq0_b  = (const float*)d_in[27];
    const float* tk0_w  = (const float*)d_in[28];
    const float* tk0_b  = (const float*)d_in[29];
    const float* tq1_w  = (const float*)d_in[30];
    const float* tq1_b  = (const float*)d_in[31];
    const float* tk1_w  = (const float*)d_in[32];
    const float* tk1_b  = (const float*)d_in[33];

    float* out = (float*)d_out;

    // ---- workspace layout (floats) ----
    float* ws     = (float*)d_ws;
    float* differ = ws;                 // NBUF
    float* Qb     = ws + 1 * NBUF;      // NBUF
    float* Kb     = ws + 2 * NBUF;      // NBUF
    float* T1     = ws + 3 * NBUF;      // NBUF  (xconv / Q', Q'')
    float* T2     = ws + 4 * NBUF;      // NBUF  (K', K'')
    float* Sb     = ws + 5 * NBUF;      // BN*65536 scores (ld padded per branch)
    float* xr     = Sb + (long long)BN * 65536;          // BN*RC*PLANE
    float* theta  = xr + (long long)BN * RC * PLANE;     // BN*RC
    float* coefs  = theta + BN * RC;                     // BN*RC*4

    const float sc_f  = 1.0f / sqrtf((float)CT);     // 1992
    const float sc_c  = 1.0f / sqrtf((float)FT);     // 14442
    const float sc_t  = 1.0f / sqrtf((float)CF);     // 464
    const float sc_t2 = 1.0f / sqrtf((float)L2OUT);  // 232

    // ---------- projections ----------
    launch_gemm(stream, w_diff, 0, CC, 1, 0,  x, CB, (int)PLANE, 1, 0, 0,
                T1, CB, (int)PLANE, CC, (int)PLANE, CC, 1.0f, b_diff, 0, BN);
    {
        long long total = NBUF;
        differ_kernel<<<(unsigned)((total + 255) / 256), 256, 0, stream>>>(x, T1, differ, total);
    }
    launch_gemm(stream, wq, 0, CC, 1, 0, differ, CB, (int)PLANE, 1, 0, 0,
                Qb, CB, (int)PLANE, CC, (int)PLANE, CC, 1.0f, bq, 0, BN);
    launch_gemm(stream, wk, 0, CC, 1, 0, differ, CB, (int)PLANE, 1, 0, 0,
                Kb, CB, (int)PLANE, CC, (int)PLANE, CC, 1.0f, bk, 0, BN);

    // ---------- residual branch ----------
    zero_kernel<<<(BN * RC + 255) / 256, 256, 0, stream>>>(theta, BN * RC);
    {
        dim3 g((unsigned)((PLANE + 255) / 256), RC, BN);
        resconv_kernel<<<g, 256, 0, stream>>>(inp, w_res, b_res, bn_g, bn_b, bn_m, bn_v, xr, theta);
    }
    coefs_kernel<<<1, 128, 0, stream>>>(theta, dy_w1, dy_b1, dy_w2, dy_b2, coefs);

    // ---------- frequency attention (N=58, d=1992) -> out channels [0,8) ----------
    float* outF = out;
    launch_gemm(stream, Qb, CB, CT, 1, 0,  Kb, CB, CT, 1, 0, 1,
                Sb, 4096, 64, FF, FF, CT, sc_f, nullptr, 0, BN);
    softmax_col<<<dim3(FF, BN), 32, 0, stream>>>(Sb, 4096, 64, FF, FF);
    launch_gemm(stream, Sb, 4096, 64, 1, 0,  differ, CB, CT, 1, 0, 0,
                outF, RC * PLANE, CT, FF, CT, FF, 1.0f, nullptr, 0, BN);
    launch_gemm(stream, fq0_w, 0, FF, 1, 0,  Qb, CB, CT, 1, 0, 0,
                T1, CB, CT, FF, CT, FF, 1.0f, fq0_b, 0, BN);
    launch_gemm(stream, fk0_w, 0, FF, 1, 0,  Kb, CB, CT, 1, 0, 0,
                T2, CB, CT, FF, CT, FF, 1.0f, fk0_b, 0, BN);
    launch_gemm(stream, T1, CB, CT, 1, 0,  T2, CB, CT, 1, 0, 1,
                Sb, 4096, 64, FF, FF, CT, sc_f, nullptr, 0, BN);
    softmax_col<<<dim3(FF, BN), 32, 0, stream>>>(Sb, 4096, 64, FF, FF);
    launch_gemm(stream, Sb, 4096, 64, 1, 0,  differ, CB, CT, 1, 0, 0,
                outF, RC * PLANE, CT, FF, CT, FF, 1.0f, nullptr, 1, BN);

    // ---------- channel attention (N=8, d=14442) -> out channels [8,16) ----------
    float* outC = out + 8 * PLANE;
    launch_gemm(stream, Qb, CB, FT, 1, 0,  Kb, CB, FT, 1, 0, 1,
                Sb, 256, 16, CC, CC, FT, sc_c, nullptr, 0, BN);
    softmax_col<<<dim3(CC, BN), 32, 0, stream>>>(Sb, 256, 16, CC, CC);
    launch_gemm(stream, Sb, 256, 16, 1, 0,  differ, CB, FT, 1, 0, 0,
                outC, RC * PLANE, FT, CC, FT, CC, 1.0f, nullptr, 0, BN);
    launch_gemm(stream, cq0_w, 0, CC, 1, 0,  Qb, CB, FT, 1, 0, 0,
                T1, CB, FT, CC, FT, CC, 1.0f, cq0_b, 0, BN);
    launch_gemm(stream, ck0_w, 0, CC, 1, 0,  Kb, CB, FT, 1, 0, 0,
                T2, CB, FT, CC, FT, CC, 1.0f, ck0_b, 0, BN);
    launch_gemm(stream, T1, CB, FT, 1, 0,  T2, CB, FT, 1, 0, 1,
                Sb, 256, 16, CC, CC, FT, sc_c, nullptr, 0, BN);
    softmax_col<<<dim3(CC, BN), 32, 0, stream>>>(Sb, 256, 16, CC, CC);
    launch_gemm(stream, Sb, 256, 16, 1, 0,  differ, CB, FT, 1, 0, 0,
                outC, RC * PLANE, FT, CC, FT, CC, 1.0f, nullptr, 1, BN);

    // ---------- time attention (N=249, d=464) -> out channels [16,24) ----------
    float* outT = out + 16 * PLANE;
    launch_gemm(stream, Qb, CB, CF, 1, 0,  Kb, CB, CF, 1, 0, 1,
                Sb, 65536, 256, TT, TT, CF, sc_t, nullptr, 0, BN);
    softmax_col<<<dim3(TT, BN), 32, 0, stream>>>(Sb, 65536, 256, TT, TT);
    launch_gemm(stream, Sb, 65536, 256, 1, 0,  differ, CB, CF, 1, 0, 0,
                outT, RC * PLANE, CF, TT, CF, TT, 1.0f, nullptr, 0, BN);
    launch_gemm(stream, tq0_w, 0, TT, 1, 0,  Qb, CB, CF, 1, 0, 0,
                T1, CB, CF, TT, CF, TT, 1.0f, tq0_b, 0, BN);
    launch_gemm(stream, tk0_w, 0, TT, 1, 0,  Kb, CB, CF, 1, 0, 0,
                T2, CB, CF, TT, CF, TT, 1.0f, tk0_b, 0, BN);
    launch_gemm(stream, T1, CB, CF, 1, 0,  T2, CB, CF, 1, 0, 1,
                Sb, 65536, 256, TT, TT, CF, sc_t, nullptr, 0, BN);
    softmax_col<<<dim3(TT, BN), 32, 0, stream>>>(Sb, 65536, 256, TT, TT);
    launch_gemm(stream, Sb, 65536, 256, 1, 0,  differ, CB, CF, 1, 0, 0,
                outT, RC * PLANE, CF, TT, CF, TT, 1.0f, nullptr, 1, BN);
    // ksize-2 stride-2 conv: out[o,j] = sum_r sum_i w[o,i,r] * Q[i, 2j+r] + b
    launch_gemm(stream, tq1_w, 0, 2 * TT, 2, 0,  Qb, CB, CF, 2, 0, 0,
                T1, CB, L2OUT, TT, L2OUT, TT, 1.0f, tq1_b, 0, BN);
    launch_gemm(stream, tq1_w, 0, 2 * TT, 2, 1,  Qb, CB, CF, 2, 1, 0,
                T1, CB, L2OUT, TT, L2OUT, TT, 1.0f, nullptr, 1, BN);
    launch_gemm(stream, tk1_w, 0, 2 * TT, 2, 0,  Kb, CB, CF, 2, 0, 0,
                T2, CB, L2OUT, TT, L2OUT, TT, 1.0f, tk1_b, 0, BN);
    launch_gemm(stream, tk1_w, 0, 2 * TT, 2, 1,  Kb, CB, CF, 2, 1, 0,
                T2, CB, L2OUT, TT, L2OUT, TT, 1.0f, nullptr, 1, BN);
    launch_gemm(stream, T1, CB, L2OUT, 1, 0,  T2, CB, L2OUT, 1, 0, 1,
                Sb, 65536, 256, TT, TT, L2OUT, sc_t2, nullptr, 0, BN);
    softmax_col<<<dim3(TT, BN), 32, 0, stream>>>(Sb, 65536, 256, TT, TT);
    launch_gemm(stream, Sb, 65536, 256, 1, 0,  differ, CB, CF, 1, 0, 0,
                outT, RC * PLANE, CF, TT, CF, TT, 1.0f, nullptr, 1, BN);

    // ---------- out += dyrelu(xr) ----------
    {
        long long total = (long long)BN * RC * PLANE;
        final_kernel<<<(unsigned)((total + 255) / 256), 256, 0, stream>>>(out, xr, coefs, total);
    }
}